// PairNorm_28819230556491
// MI455X (gfx1250) — compile-verified
//
#include <hip/hip_runtime.h>

// PairNorm for MI455X / gfx1250 (wave32).
//   Pass 1: per-segment sum(x) via V_WMMA_F32_16X16X4_F32 (exact f32 column
//           sums, all-ones A matrix), plus sum(||x||^2) and counts.
//           Sorted batch -> 32-row-group uniformity fast path (no per-tile
//           segment checks in steady state).
//   Finalize: mean[s][d], inv_denom[s] = rsqrt(S2/c - ||m||^2).
//   Pass 2: out = (x - mean[seg]) * inv_denom[seg]; reverse traversal to hit
//           the L2-resident tail of x left by pass 1; NT stores for out.

#define D 64
#define NSEG 512

typedef float v2f __attribute__((ext_vector_type(2)));
typedef float v4f __attribute__((ext_vector_type(4)));
typedef float v8f __attribute__((ext_vector_type(8)));

__global__ void pairnorm_pass1(const float* __restrict__ x,
                               const long long* __restrict__ batch,
                               float* __restrict__ sums,   // [NSEG][D]
                               float* __restrict__ cnt,    // [NSEG]
                               float* __restrict__ sumsq,  // [NSEG]
                               int N) {
    const int tid   = threadIdx.x;
    const int lane  = tid & 31;
    const int wib   = tid >> 5;
    const int wavesPerBlk = blockDim.x >> 5;
    const int W     = blockIdx.x * wavesPerBlk + wib;
    const int cg    = W & 3;          // 16-column group: 0..3
    const int chunk = W >> 2;         // contiguous row chunk per 4-wave set
    const int numChunks = (gridDim.x * wavesPerBlk) >> 2;

    int rowsPerChunk = ((N + numChunks - 1) / numChunks + 31) & ~31;
    int rowStart = chunk * rowsPerChunk;
    int rowEnd   = rowStart + rowsPerChunk;
    if (rowEnd > N) rowEnd = N;

    const int rOff = (lane < 16) ? 0 : 2;            // my first row within a 4-row tile
    const int col  = cg * 16 + (lane & 15);          // my column (0..63)

    v8f c0, c1;
    #pragma unroll
    for (int i = 0; i < 8; ++i) { c0[i] = 0.0f; c1[i] = 0.0f; }
    v2f ones; ones.x = 1.0f; ones.y = 1.0f;          // A = 16x4 all-ones (f32 exact)

    float sqAcc  = 0.0f;   // per-lane sum of squares (my 16 columns only)
    int   cntAcc = 0;      // rows accumulated (consumed by cg==0 only)
    int   curSeg = -1;

    auto flush = [&]() {
        if (curSeg >= 0) {
            // D row M=0 lives in VGPR0 (c[0]), lanes 0-15, N = lane.
            float colsum = c0[0] + c1[0];
            if (lane < 16) atomicAdd(&sums[curSeg * D + cg * 16 + lane], colsum);
            float r = sqAcc;
            #pragma unroll
            for (int off = 16; off > 0; off >>= 1) r += __shfl_xor(r, off, 32);
            if (lane == 0) {
                atomicAdd(&sumsq[curSeg], r);
                if (cg == 0) atomicAdd(&cnt[curSeg], (float)cntAcc);
            }
            #pragma unroll
            for (int i = 0; i < 8; ++i) { c0[i] = 0.0f; c1[i] = 0.0f; }
            sqAcc = 0.0f; cntAcc = 0; curSeg = -1;
        }
    };

    for (int g = rowStart; g < rowEnd; g += 32) {
        int gEnd = g + 32;
        if (gEnd > rowEnd) gEnd = rowEnd;
        const bool fullG = (gEnd - g) == 32;

        // Endpoint segment ids for the group (sorted batch => uniform iff equal).
        int sF = 0, sB = 0;
        if (lane == 0) { sF = (int)batch[g]; sB = (int)batch[gEnd - 1]; }
        sF = __shfl(sF, 0, 32);
        sB = __shfl(sB, 0, 32);

        if (fullG && sF == sB) {
            // --- Fast path: whole 32-row group in one segment. ---
            if (sF != curSeg) { flush(); curSeg = sF; }
            const float* p = x + (g + rOff) * D + col;
            #pragma unroll
            for (int t = 0; t < 8; ++t) {
                v2f b;
                b.x = p[t * 4 * D];          // B row K=rOff
                b.y = p[t * 4 * D + D];      // B row K=rOff+1
                if (t & 1)
                    c1 = __builtin_amdgcn_wmma_f32_16x16x4_f32(
                            false, ones, false, b, (short)0, c1, false, false);
                else
                    c0 = __builtin_amdgcn_wmma_f32_16x16x4_f32(
                            false, ones, false, b, (short)0, c0, false, false);
                sqAcc += b.x * b.x + b.y * b.y;
            }
            cntAcc += 32;
        } else {
            // --- Slow path: segment boundary or ragged tail inside group. ---
            for (int base = g; base < gEnd; base += 4) {
                const bool full = (base + 4 <= gEnd);
                int sL = 0;
                if (lane < 4 && base + lane < gEnd) sL = (int)batch[base + lane];
                int s0 = __shfl(sL, 0, 32);
                int s1 = __shfl(sL, 1, 32);
                int s2 = __shfl(sL, 2, 32);
                int s3 = __shfl(sL, 3, 32);

                if (full && s0 == s1 && s1 == s2 && s2 == s3) {
                    if (s0 != curSeg) { flush(); curSeg = s0; }
                    const int rA = base + rOff;
                    v2f b;
                    b.x = x[rA * D + col];
                    b.y = x[(rA + 1) * D + col];
                    c0 = __builtin_amdgcn_wmma_f32_16x16x4_f32(
                            false, ones, false, b, (short)0, c0, false, false);
                    sqAcc  += b.x * b.x + b.y * b.y;
                    cntAcc += 4;
                } else {
                    flush();
                    const int rA = base + rOff;
                    if (rA < gEnd) {
                        int s = (lane < 16) ? s0 : s2;
                        float v = x[rA * D + col];
                        atomicAdd(&sums[s * D + col], v);
                        atomicAdd(&sumsq[s], v * v);
                    }
                    const int rB = rA + 1;
                    if (rB < gEnd) {
                        int s = (lane < 16) ? s1 : s3;
                        float v = x[rB * D + col];
                        atomicAdd(&sums[s * D + col], v);
                        atomicAdd(&sumsq[s], v * v);
                    }
                    if (cg == 0 && lane < 4 && base + lane < gEnd)
                        atomicAdd(&cnt[sL], 1.0f);
                }
            }
        }
    }
    flush();
}

__global__ void pairnorm_finalize(const float* __restrict__ sums,
                                  const float* __restrict__ cnt,
                                  const float* __restrict__ sumsq,
                                  float* __restrict__ mean,
                                  float* __restrict__ invden) {
    __shared__ float red[D];
    const int s = blockIdx.x;
    const int d = threadIdx.x;
    const float invc = 1.0f / fmaxf(cnt[s], 1.0f);
    const float m = sums[s * D + d] * invc;
    mean[s * D + d] = m;
    red[d] = m * m;
    __syncthreads();
    if (d == 0) {
        float msq = 0.0f;
        #pragma unroll
        for (int i = 0; i < D; ++i) msq += red[i];
        float var = fmaxf(sumsq[s] * invc - msq, 1e-12f);
        invden[s] = rsqrtf(var);
    }
}

__global__ void pairnorm_apply(const float* __restrict__ x,
                               const long long* __restrict__ batch,
                               const float* __restrict__ mean,
                               const float* __restrict__ invden,
                               float* __restrict__ out,
                               int N) {
    const int total = N * (D / 4);                 // float4 units
    const v4f* __restrict__ x4 = (const v4f*)x;
    const v4f* __restrict__ m4 = (const v4f*)mean;
    v4f* __restrict__ o4 = (v4f*)out;
    const int stride = gridDim.x * blockDim.x;
    for (int j = blockIdx.x * blockDim.x + threadIdx.x; j < total; j += stride) {
        const int i   = total - 1 - j;             // reverse: hit L2-resident tail first
        const int row = i >> 4;                    // D/4 == 16 quads per row
        const int q   = i & 15;
        const int s   = (int)batch[row];
        v4f xv = __builtin_nontemporal_load(&x4[i]);   // x is dead after this pass
        v4f mv = m4[s * 16 + q];
        float inv = invden[s];
        v4f o;
        o.x = (xv.x - mv.x) * inv;
        o.y = (xv.y - mv.y) * inv;
        o.z = (xv.z - mv.z) * inv;
        o.w = (xv.w - mv.w) * inv;
        __builtin_nontemporal_store(o, &o4[i]);        // out never re-read
    }
}

extern "C" void kernel_launch(void* const* d_in, const int* in_sizes, int n_in,
                              void* d_out, int out_size, void* d_ws, size_t ws_size,
                              hipStream_t stream) {
    const float*     x     = (const float*)d_in[0];
    const long long* batch = (const long long*)d_in[1];
    const int N = in_sizes[1];

    float* ws     = (float*)d_ws;
    float* sums   = ws;                    // NSEG*D
    float* cnt    = sums + NSEG * D;       // NSEG
    float* sumsq  = cnt + NSEG;            // NSEG
    float* mean   = sumsq + NSEG;          // NSEG*D
    float* invden = mean + NSEG * D;       // NSEG

    // Zero the accumulators (sums + cnt + sumsq).
    hipMemsetAsync(ws, 0, (size_t)(NSEG * D + 2 * NSEG) * sizeof(float), stream);

    pairnorm_pass1<<<2048, 256, 0, stream>>>(x, batch, sums, cnt, sumsq, N);
    pairnorm_finalize<<<NSEG, D, 0, stream>>>(sums, cnt, sumsq, mean, invden);
    pairnorm_apply<<<4096, 256, 0, stream>>>(x, batch, mean, invden, (float*)d_out, N);
}